// Corr1d_x_group_51951924412916
// MI455X (gfx1250) — compile-verified
//
#include <hip/hip_runtime.h>
#include <hip/hip_bf16.h>

typedef __attribute__((ext_vector_type(2))) float v2f;
typedef __attribute__((ext_vector_type(8))) float v8f;

namespace {
constexpr int kN = 8;
constexpr int kG = 4;          // groups; 4 channels per group
constexpr int kH = 256;
constexpr int kW = 512;
constexpr int kTop = 27;       // displacement channels, s = ch - 23 in [-23, 3]
constexpr int kWavesPerBlock = 4;   // 128 threads, 32 KB LDS / block
constexpr long kHW = (long)kH * kW;
}

// One wave computes a 32-wide w strip for one (n, g, h):
//   2 A-tiles (16 w-rows x 4 chans) x 4 B-tiles each -> 8x V_WMMA_F32_16X16X4_F32
//   banded 16x64 panels staged in LDS, diagonals extracted, coalesced NT stores.
__global__ __launch_bounds__(128) void corr1d_group_wmma_kernel(
    const float* __restrict__ f1g,   // l_in  [8,16,256,512]
    const float* __restrict__ f2g,   // r_in  [8,16,256,512]
    float* __restrict__ out) {       // [8,108,256,512]
  const int lane = threadIdx.x & 31;
  const int wave = threadIdx.x >> 5;
  const int half = lane >> 4;        // which 16-lane half
  const int ln   = lane & 15;

  const int tile = blockIdx.x * kWavesPerBlock + wave;  // 131072 wave-tiles total
  const int WT = kW / 32;            // 16 strips per row
  const int wt = tile & (WT - 1);
  const int t1 = tile / WT;
  const int h  = t1 & (kH - 1);
  const int t2 = t1 / kH;
  const int g  = t2 & (kG - 1);
  const int n  = t2 >> 2;
  const int w0 = wt * 32;

  // channel-plane base pointers: element (n, 4g+c, h, w)
  const float* f1b = f1g + (long)(n * 16 + g * 4) * kHW + (long)h * kW;
  const float* f2b = f2g + (long)(n * 16 + g * 4) * kHW + (long)h * kW;
  const int c0 = half * 2;           // this half-wave supplies K = c0, c0+1

  __shared__ float lds[kWavesPerBlock][2][16][64];

  // ---- load 5 B-tiles (4 x 16 fp32 each), col origins w0-32+16t ----
  // B layout (mirrors documented A/sparse-B striping): lanes 0-15 rows K=0,1;
  // lanes 16-31 rows K=2,3; N = lane%16 within one VGPR.
  v2f B[5];
#pragma unroll
  for (int t = 0; t < 5; ++t) {
    const int wc = w0 - 32 + 16 * t + ln;
    float b0 = 0.0f, b1 = 0.0f;
    if (wc >= 0 && wc < kW) {        // zero-pad semantics of the reference
      b0 = f2b[(long)c0 * kHW + wc];
      b1 = f2b[(long)(c0 + 1) * kHW + wc];
    }
    B[t].x = b0;
    B[t].y = b1;
  }

  // ---- 2 row-blocks x 4 WMMAs; dump each 16x16 D into this wave's LDS panel ----
#pragma unroll
  for (int mt = 0; mt < 2; ++mt) {
    const int wA = w0 + 16 * mt + ln;    // A row M = ln  ->  w = w0 + 16*mt + M
    v2f A;
    A.x = f1b[(long)c0 * kHW + wA];      // documented 16x4 fp32 A layout
    A.y = f1b[(long)(c0 + 1) * kHW + wA];
#pragma unroll
    for (int tt = 0; tt < 4; ++tt) {     // panel cols [16*tt, 16*tt+16), base w0-32+16*mt
      v8f Cz = {};
      v8f D = __builtin_amdgcn_wmma_f32_16x16x4_f32(
          /*neg_a=*/false, A, /*neg_b=*/false, B[mt + tt],
          /*c_mod=*/(short)0, Cz, /*reuse_a=*/false, /*reuse_b=*/false);
      // D element (row r + 8*half, col ln) per 16x16 f32 C/D layout
#pragma unroll
      for (int r = 0; r < 8; ++r)
        lds[wave][mt][r + 8 * half][tt * 16 + ln] = D[r];
    }
  }
  // Same-wave DS ops are in-order (DScnt); no barrier needed — LDS regions are
  // private per wave and reads below depend on this wave's stores only.

  // ---- extract 27 band diagonals; lane owns w = w0 + lane ----
  // panel(mt)[M][col] = corr(w0+16*mt+M, (w0-32+16*mt)+col)  =>  col = ln + s + 32
  float* outb = out + (long)(n * (kG * kTop) + g * kTop) * kHW
                    + (long)h * kW + w0 + lane;
#pragma unroll
  for (int ch = 0; ch < kTop; ++ch) {
    const int s = ch - 23;                                // in [-23, 3]
    const float v = lds[wave][half][ln][ln + s + 32] * 0.25f;  // /sumelems(=4)
    // 453 MB output stream > 192 MB L2: bypass with non-temporal store
    __builtin_nontemporal_store(v, outb + (long)ch * kHW);
  }
}

extern "C" void kernel_launch(void* const* d_in, const int* in_sizes, int n_in,
                              void* d_out, int out_size, void* d_ws, size_t ws_size,
                              hipStream_t stream) {
  const float* l_in = (const float*)d_in[0];
  const float* r_in = (const float*)d_in[1];
  float* out = (float*)d_out;

  const int waveTiles = kN * kG * kH * (kW / 32);        // 131072
  const int blocks = waveTiles / kWavesPerBlock;         // 32768
  corr1d_group_wmma_kernel<<<blocks, kWavesPerBlock * 32, 0, stream>>>(l_in, r_in, out);
}